// dgl_gat_1099511628217
// MI455X (gfx1250) — compile-verified
//
#include <hip/hip_runtime.h>
#include <math.h>

typedef __attribute__((ext_vector_type(2))) float v2f;
typedef __attribute__((ext_vector_type(8))) float v8f;

// ---------------------------------------------------------------------------
// GEMM C[M,Ncol] = A[M,K] @ B[K,Ncol], fp32 via V_WMMA_F32_16X16X4_F32.
// One wave computes one 16x16 output tile; 8 waves (256 thr) per block share
// the same tileN column strip so B fragments hit L1/L2.
// A-frag layout (ISA 7.12.2, 32-bit A 16x4): lanes 0-15 -> K={0,1},
// lanes 16-31 -> K={2,3}. C/D layout: VGPR i -> M=i (+8 for upper half-wave).
// ---------------------------------------------------------------------------
__global__ __launch_bounds__(256) void gemm_wmma_f32(
    const float* __restrict__ A, const float* __restrict__ B,
    float* __restrict__ C, int M, int K, int Ncol)
{
    const int lane  = threadIdx.x & 31;
    const int wave  = threadIdx.x >> 5;
    const int tileM = blockIdx.x * 8 + wave;       // wave-uniform
    const int tileN = blockIdx.y;
    if (tileM * 16 >= M) return;                   // whole wave exits: EXEC stays all-1s
    const int r     = lane & 15;
    const int kHalf = lane >> 4;                   // 0 -> K pair {0,1}, 1 -> {2,3}

    const float* Arow = A + (size_t)(tileM * 16 + r) * K + kHalf * 2;
    const float* Bcol = B + (size_t)(kHalf * 2) * Ncol + (tileN * 16 + r);

    v8f acc = {};
#pragma unroll 4
    for (int k = 0; k < K; k += 4) {
        v2f a = *(const v2f*)(Arow + k);           // 8B-aligned contiguous pair
        v2f b;
        b.x = Bcol[(size_t)k * Ncol];
        b.y = Bcol[(size_t)k * Ncol + Ncol];
        acc = __builtin_amdgcn_wmma_f32_16x16x4_f32(
            /*neg_a=*/false, a, /*neg_b=*/false, b,
            /*c_mod=*/(short)0, acc, /*reuse_a=*/false, /*reuse_b=*/false);
    }

    float* Crow = C + (size_t)(tileM * 16 + kHalf * 8) * Ncol + tileN * 16 + r;
#pragma unroll
    for (int i = 0; i < 8; ++i)
        Crow[(size_t)i * Ncol] = acc[i];
}

// el[n,h] = <h[n,h,:], al[h,:]> ; er likewise. One thread per (node,head).
__global__ __launch_bounds__(256) void node_attn_dots(
    const float* __restrict__ h, const float* __restrict__ al,
    const float* __restrict__ ar, float* __restrict__ el,
    float* __restrict__ er, int n, int H, int Dh)
{
    int t = blockIdx.x * blockDim.x + threadIdx.x;
    if (t >= n * H) return;
    int hd = t % H;
    const float* hp  = h  + (size_t)t * Dh;        // h is [n,H,Dh]; t = node*H + hd
    const float* alp = al + hd * Dh;
    const float* arp = ar + hd * Dh;
    float sl = 0.f, sr = 0.f;
#pragma unroll 8
    for (int d = 0; d < Dh; ++d) { float v = hp[d]; sl += v * alp[d]; sr += v * arp[d]; }
    el[t] = sl; er[t] = sr;
}

// Monotonic float<->uint mapping so unsigned atomicMax == float max.
__device__ __forceinline__ unsigned fmap(float f) {
    unsigned u = __float_as_uint(f);
    return (u & 0x80000000u) ? ~u : (u | 0x80000000u);
}
__device__ __forceinline__ float funmap(unsigned m) {
    unsigned u = (m & 0x80000000u) ? (m & 0x7FFFFFFFu) : ~m;
    return __uint_as_float(u);
}

__global__ __launch_bounds__(256) void edge_max(
    const int* __restrict__ src, const int* __restrict__ dst,
    const float* __restrict__ el, const float* __restrict__ er,
    unsigned* __restrict__ mmax, int E, int H, float slope)
{
    int e = blockIdx.x * blockDim.x + threadIdx.x;
    if (e >= E) return;
    int s = src[e], d = dst[e];
    for (int h = 0; h < H; ++h) {
        float v = el[s * H + h] + er[d * H + h];
        v = v > 0.f ? v : slope * v;
        atomicMax(&mmax[d * H + h], fmap(v));
    }
}

__global__ __launch_bounds__(256) void edge_denom(
    const int* __restrict__ src, const int* __restrict__ dst,
    const float* __restrict__ el, const float* __restrict__ er,
    const unsigned* __restrict__ mmax, float* __restrict__ denom,
    int E, int H, float slope)
{
    int e = blockIdx.x * blockDim.x + threadIdx.x;
    if (e >= E) return;
    int s = src[e], d = dst[e];
    for (int h = 0; h < H; ++h) {
        float v = el[s * H + h] + er[d * H + h];
        v = v > 0.f ? v : slope * v;
        float m = funmap(mmax[d * H + h]);
        unsafeAtomicAdd(&denom[d * H + h], __expf(v - m));
    }
}

// One wave per edge; lane = feature dim. Gathers h[src] (L2-resident on
// MI455X: 51 MB << 192 MB L2) and scatter-adds into out[dst].
__global__ __launch_bounds__(256) void edge_agg(
    const int* __restrict__ src, const int* __restrict__ dst,
    const float* __restrict__ el, const float* __restrict__ er,
    const unsigned* __restrict__ mmax, const float* __restrict__ denom,
    const float* __restrict__ hfeat, float* __restrict__ out,
    int E, int H, int Dh, float slope)
{
    int lane = threadIdx.x & 31;
    int e = blockIdx.x * 8 + (threadIdx.x >> 5);
    if (e >= E || lane >= Dh) return;
    int s = src[e], d = dst[e];
    for (int h = 0; h < H; ++h) {
        float v = el[s * H + h] + er[d * H + h];
        v = v > 0.f ? v : slope * v;
        float m = funmap(mmax[d * H + h]);
        float alpha = __expf(v - m) / denom[d * H + h];
        float val = hfeat[((size_t)s * H + h) * Dh + lane] * alpha;
        unsafeAtomicAdd(&out[((size_t)d * H + h) * Dh + lane], val);
    }
}

__global__ __launch_bounds__(256) void bias_elu(
    const float* __restrict__ in, const float* __restrict__ b,
    float* __restrict__ out, long total, int F)
{
    long t = blockIdx.x * (long)blockDim.x + threadIdx.x;
    if (t >= total) return;
    float v = in[t] + b[t % F];
    out[t] = v > 0.f ? v : expm1f(v);              // ELU alpha=1
}

extern "C" void kernel_launch(void* const* d_in, const int* in_sizes, int n_in,
                              void* d_out, int out_size, void* d_ws, size_t ws_size,
                              hipStream_t stream) {
    const float* x   = (const float*)d_in[0];
    const int*   src = (const int*)d_in[1];
    const int*   dst = (const int*)d_in[2];
    const float* W1  = (const float*)d_in[3];
    const float* al1 = (const float*)d_in[4];
    const float* ar1 = (const float*)d_in[5];
    const float* b1  = (const float*)d_in[6];
    const float* W2  = (const float*)d_in[7];
    const float* al2 = (const float*)d_in[8];
    const float* ar2 = (const float*)d_in[9];
    const float* b2  = (const float*)d_in[10];
    float* out = (float*)d_out;

    const int N = 50000, E = 800000, F = 256, H1 = 8, D1 = 32, C = 16;

    char* w = (char*)d_ws;
    size_t off = 0;
    auto give = [&](size_t bytes) -> char* {
        char* p = w + off; off += (bytes + 255) & ~(size_t)255; return p;
    };
    float*    h1  = (float*)give((size_t)N * F * 4);
    float*    x1  = (float*)give((size_t)N * F * 4);
    float*    el1 = (float*)give((size_t)N * H1 * 4);
    float*    er1 = (float*)give((size_t)N * H1 * 4);
    unsigned* m1  = (unsigned*)give((size_t)N * H1 * 4);
    float*    dn1 = (float*)give((size_t)N * H1 * 4);
    float*    h2  = (float*)give((size_t)N * C * 4);
    float*    o2  = (float*)give((size_t)N * C * 4);
    float*    el2 = (float*)give((size_t)N * 4);
    float*    er2 = (float*)give((size_t)N * 4);
    unsigned* m2  = (unsigned*)give((size_t)N * 4);
    float*    dn2 = (float*)give((size_t)N * 4);

    // zero accumulators (sentinel 0 is the floor of the monotonic-uint map)
    hipMemsetAsync(x1, 0, (size_t)N * F * 4, stream);
    hipMemsetAsync(m1, 0, (size_t)N * H1 * 4, stream);
    hipMemsetAsync(dn1, 0, (size_t)N * H1 * 4, stream);
    hipMemsetAsync(o2, 0, (size_t)N * C * 4, stream);
    hipMemsetAsync(m2, 0, (size_t)N * 4, stream);
    hipMemsetAsync(dn2, 0, (size_t)N * 4, stream);

    // ---- layer 1 (neg_slope = 1.0 -> LeakyReLU is identity, kept generic)
    { dim3 g((N / 16 + 7) / 8, F / 16);
      gemm_wmma_f32<<<g, 256, 0, stream>>>(x, W1, h1, N, F, F); }
    node_attn_dots<<<(N * H1 + 255) / 256, 256, 0, stream>>>(h1, al1, ar1, el1, er1, N, H1, D1);
    edge_max  <<<(E + 255) / 256, 256, 0, stream>>>(src, dst, el1, er1, m1, E, H1, 1.0f);
    edge_denom<<<(E + 255) / 256, 256, 0, stream>>>(src, dst, el1, er1, m1, dn1, E, H1, 1.0f);
    edge_agg  <<<(E + 7) / 8,     256, 0, stream>>>(src, dst, el1, er1, m1, dn1, h1, x1, E, H1, D1, 1.0f);
    bias_elu  <<<(int)(((long)N * F + 255) / 256), 256, 0, stream>>>(x1, b1, x1, (long)N * F, F);

    // ---- layer 2 (H=1, C=16, neg_slope = 0.2)
    { dim3 g((N / 16 + 7) / 8, C / 16);
      gemm_wmma_f32<<<g, 256, 0, stream>>>(x1, W2, h2, N, F, C); }
    node_attn_dots<<<(N + 255) / 256, 256, 0, stream>>>(h2, al2, ar2, el2, er2, N, 1, C);
    edge_max  <<<(E + 255) / 256, 256, 0, stream>>>(src, dst, el2, er2, m2, E, 1, 0.2f);
    edge_denom<<<(E + 255) / 256, 256, 0, stream>>>(src, dst, el2, er2, m2, dn2, E, 1, 0.2f);
    edge_agg  <<<(E + 7) / 8,     256, 0, stream>>>(src, dst, el2, er2, m2, dn2, h2, o2, E, 1, C, 0.2f);
    bias_elu  <<<(int)(((long)N * C + 255) / 256), 256, 0, stream>>>(o2, b2, out, (long)N * C, C);
}